// GaussianAttention_89910845374658
// MI455X (gfx1250) — compile-verified
//
#include <hip/hip_runtime.h>
#include <hip/hip_bf16.h>
#include <cstdint>

typedef __attribute__((ext_vector_type(2))) float v2f;
typedef __attribute__((ext_vector_type(8))) float v8f;

#define FILTER_SIZE 21
#define PAD 10
#define D_DIM 1024
#define T_DIM 8192
#define B_DIM 4096
#define MASK_VALUE 1e-8f
#define HALO 12   // 12-float lead pad keeps LDS b128 stores 16B-aligned

__device__ __forceinline__ float sigmoidf(float x) {
    return 1.0f / (1.0f + __expf(-x));
}

// ---------------------------------------------------------------------------
// Kernel A: p = sigmoid(query @ proj_w.T + proj_b) via V_WMMA_F32_16X16X4_F32,
// then build the normalized 21-tap per-row filter into ker_out (B x 21).
// One wave (32 threads) per 16-row tile of query. B-fragment is built
// branch-free (clamped row + 0/1 scale) to avoid per-iteration EXEC churn.
// ---------------------------------------------------------------------------
__global__ __launch_bounds__(32)
void ga_proj_filter_kernel(const float* __restrict__ query,
                           const float* __restrict__ proj_w,
                           const float* __restrict__ proj_b,
                           float* __restrict__ ker_out) {
    const int  lane    = threadIdx.x;      // 0..31
    const int  m       = lane & 15;        // A: row-in-tile; B: N column index
    const bool hi      = lane >= 16;       // hi half carries K={2,3}
    const int  rowBase = blockIdx.x * 16;

    const float* qrow   = query + (size_t)(rowBase + m) * D_DIM;
    const float* brow   = proj_w + (size_t)(m & 3) * D_DIM;  // clamped, uniform
    const float  bscale = (m < 4) ? 1.0f : 0.0f;             // zero N=4..15
    const int    kshift = hi ? 2 : 0;

    v8f acc = {0.f, 0.f, 0.f, 0.f, 0.f, 0.f, 0.f, 0.f};

    for (int k0 = 0; k0 < D_DIM; k0 += 4) {
        const int kk = k0 + kshift;
        // A fragment (16x4 f32): lane m holds A[m][kk], A[m][kk+1]
        v2f a;
        a.x = qrow[kk];
        a.y = qrow[kk + 1];
        // B fragment (4x16 f32): B[k][n] = proj_w[n][k] for n<4, else 0
        v2f b;
        b.x = brow[kk] * bscale;
        b.y = brow[kk + 1] * bscale;
        acc = __builtin_amdgcn_wmma_f32_16x16x4_f32(false, a, false, b,
                                                    (short)0, acc, false, false);
    }

    // C layout: acc[j] = C[M=j][N=lane] (lanes 0-15), C[M=j+8][N=lane-16] (hi).
    // Only columns N=0..3 are meaningful -> park them in LDS.
    __shared__ float psh[16 * 4];
    if (m < 4) {
        #pragma unroll
        for (int j = 0; j < 8; ++j)
            psh[(j + (hi ? 8 : 0)) * 4 + m] = acc[j];
    }
    __syncthreads();

    if (lane < 16) {
        const float p0 = sigmoidf(psh[lane * 4 + 0] + proj_b[0]);  // mu
        const float p1 = sigmoidf(psh[lane * 4 + 1] + proj_b[1]);  // alpha
        const float p2 = sigmoidf(psh[lane * 4 + 2] + proj_b[2]);  // sigma0
        const float p3 = sigmoidf(psh[lane * 4 + 3] + proj_b[3]);  // sigma1 inc

        const float mu    = (float)PAD - 2.0f * p0;  // PRIOR_TOKENS_PER_FRAME=1
        const float alpha = p1;
        const float s0    = 0.2f + p2;               // MIN_SIGMA + cumsum
        const float s1    = 0.2f + p2 + p3;
        const float i2s0  = 1.0f / (2.0f * s0), r0 = 1.0f / s0;
        const float i2s1  = 1.0f / (2.0f * s1), r1 = 1.0f / s1;

        float taps[FILTER_SIZE];
        float sum = 0.f;
        #pragma unroll
        for (int k = 0; k < FILTER_SIZE; ++k) {
            const float d  = (float)k - mu;
            const float e0 = d * i2s0;
            const float e1 = d * i2s1;
            const float g0 = __expf(-e0 * e0) * r0;
            const float g1 = __expf(-e1 * e1) * r1;
            const float v  = (1.0f + alpha) * g0 - alpha * g1;
            taps[k] = v;
            sum += v;
        }
        const float inv = 1.0f / sum;
        float* o = ker_out + (size_t)(rowBase + lane) * FILTER_SIZE;
        #pragma unroll
        for (int k = 0; k < FILTER_SIZE; ++k) o[k] = taps[k] * inv;
    }
}

// ---------------------------------------------------------------------------
// Kernel B: per-row depthwise conv (21 taps, zero-padded), mask/clip, and
// row-sum renormalization in a single HBM pass. One 256-thread block per row;
// the whole 8192-float row is staged in LDS with halos.
// ---------------------------------------------------------------------------
__global__ __launch_bounds__(256)
void ga_conv_norm_kernel(const float* __restrict__ aw,
                         const uint8_t* __restrict__ mask,
                         const float* __restrict__ ker,
                         float* __restrict__ out) {
    const int b   = blockIdx.x;
    const int tid = threadIdx.x;

    __shared__ float srow[HALO + T_DIM + 12];   // 8216 floats = 32.9 KB
    __shared__ float kc[FILTER_SIZE];
    __shared__ float wsum[8];

    // Zero halos (left: srow[0..11], right: srow[8204..8215]).
    if (tid < HALO) srow[tid] = 0.f;
    if (tid >= 32 && tid < 32 + 12) srow[HALO + T_DIM + (tid - 32)] = 0.f;
    if (tid < FILTER_SIZE) kc[tid] = ker[(size_t)b * FILTER_SIZE + tid];

    // Stage the row: coalesced float4 loads, 16B-aligned ds_store_b128.
    const float4* arow4 = (const float4*)(aw + (size_t)b * T_DIM);
    float4*       srow4 = (float4*)(srow + HALO);
    for (int i = tid; i < T_DIM / 4; i += 256) srow4[i] = arow4[i];
    __syncthreads();

    // Each thread owns t = tid + j*256 -> conflict-free LDS, coalesced stores.
    const uint8_t* mrow = mask + (size_t)b * T_DIM;
    float vals[32];
    float local = 0.f;
    #pragma unroll
    for (int j = 0; j < 32; ++j) {
        const int    t = tid + j * 256;
        const float* s = srow + 2 + t;          // srow[2+t+k] == aw[t+k-10]
        float a = 0.f;
        #pragma unroll
        for (int k = 0; k < FILTER_SIZE; ++k) a = fmaf(kc[k], s[k], a);
        float v = mrow[t] ? a : MASK_VALUE;
        v = fmaxf(v, MASK_VALUE);
        vals[j] = v;
        local += v;
    }

    // Row-sum reduction: wave32 shuffle reduce, then cross-wave via LDS.
    #pragma unroll
    for (int off = 16; off > 0; off >>= 1) local += __shfl_xor(local, off, 32);
    if ((tid & 31) == 0) wsum[tid >> 5] = local;
    __syncthreads();
    float total = 0.f;
    #pragma unroll
    for (int w = 0; w < 8; ++w) total += wsum[w];
    const float inv = 1.0f / total;

    float* orow = out + (size_t)b * T_DIM;
    #pragma unroll
    for (int j = 0; j < 32; ++j) orow[tid + j * 256] = vals[j] * inv;
}

// ---------------------------------------------------------------------------
extern "C" void kernel_launch(void* const* d_in, const int* in_sizes, int n_in,
                              void* d_out, int out_size, void* d_ws, size_t ws_size,
                              hipStream_t stream) {
    (void)in_sizes; (void)n_in; (void)out_size; (void)ws_size;
    const float*   query  = (const float*)d_in[0];
    const float*   attw   = (const float*)d_in[1];
    const uint8_t* mask   = (const uint8_t*)d_in[2];
    const float*   proj_w = (const float*)d_in[3];
    const float*   proj_b = (const float*)d_in[4];
    float*         out    = (float*)d_out;
    float*         kerbuf = (float*)d_ws;   // B*21 floats = 344 KB scratch

    ga_proj_filter_kernel<<<B_DIM / 16, 32, 0, stream>>>(query, proj_w, proj_b, kerbuf);
    ga_conv_norm_kernel<<<B_DIM, 256, 0, stream>>>(attw, mask, kerbuf, out);
}